// CRF_68710886801741
// MI455X (gfx1250) — compile-verified
//
#include <hip/hip_runtime.h>
#include <hip/hip_bf16.h>

typedef __attribute__((ext_vector_type(2))) float v2f;
typedef __attribute__((ext_vector_type(8))) float v8f;

#define TAGSET    16
#define START_TAG 14
#define STOP_TAG  15
#define SEQ_LEN   32768
#define LSTEPS    64
#define NBLK      (SEQ_LEN / LSTEPS)   // 512 scan blocks
#define FOLD      16
#define NGRP      (NBLK / FOLD)        // 32 groups == 32 waves == 1024 threads
#define GOLD_BLKS 16
#define LN2F      0.69314718055994531f

// ws layout (floats):
#define WS_MAT    0                    // 512*256 : block composites (exp-domain)
#define WS_SCALE  (NBLK * 256)         // 512     : block log-scales (nats)
#define WS_GOLD   (WS_SCALE + NBLK)    // 16      : gold partial sums

// ---- fixed cross-lane permutes as single VALU ops ----
__device__ __forceinline__ float permx16_f(float x) {          // lane ^= 16
    return __int_as_float(__builtin_amdgcn_permlanex16(
        __float_as_int(x), __float_as_int(x), 0x76543210u, 0xfedcba98u, false, false));
}
__device__ __forceinline__ float permlane16_f(float x, unsigned s0, unsigned s1) {
    return __int_as_float(__builtin_amdgcn_permlane16(
        __float_as_int(x), __float_as_int(x), s0, s1, false, false));
}
__device__ __forceinline__ float rfl_f(float x) {              // uniform broadcast of lane 0
    return __int_as_float(__builtin_amdgcn_readfirstlane(__float_as_int(x)));
}
#define XOR1_LO 0x67452301u
#define XOR1_HI 0xEFCDAB89u
#define XOR2_LO 0x54761032u
#define XOR2_HI 0xDCFE98BAu
#define XOR4_LO 0x32107654u
#define XOR4_HI 0xBA98FEDCu
#define XOR8_LO 0xFEDCBA98u
#define XOR8_HI 0x76543210u

__device__ __forceinline__ float wave_sum32(float a) {
    a += permx16_f(a);
    a += permlane16_f(a, XOR8_LO, XOR8_HI);
    a += permlane16_f(a, XOR4_LO, XOR4_HI);
    a += permlane16_f(a, XOR2_LO, XOR2_HI);
    a += permlane16_f(a, XOR1_LO, XOR1_HI);
    return a;
}

// Form the four 4x16 B chunks (rows of P) from C/D layout registers.
__device__ __forceinline__ void make_b(const float c[8], bool lo,
                                       v2f& b0, v2f& b1, v2f& b2, v2f& b3) {
    float sx[8];
    #pragma unroll
    for (int r = 0; r < 8; ++r) sx[r] = permx16_f(c[r]);
    b0.x = lo ? c[0]  : sx[2];  b0.y = lo ? c[1]  : sx[3];   // rows 0..3
    b1.x = lo ? c[4]  : sx[6];  b1.y = lo ? c[5]  : sx[7];   // rows 4..7
    b2.x = lo ? sx[0] : c[2];   b2.y = lo ? sx[1] : c[3];    // rows 8..11
    b3.x = lo ? sx[4] : c[6];   b3.y = lo ? sx[5] : c[7];    // rows 12..15
}

__device__ __forceinline__ v8f wmma16(const v2f a[4], v2f b0, v2f b1, v2f b2, v2f b3) {
    v8f acc = {0.f,0.f,0.f,0.f,0.f,0.f,0.f,0.f};
    acc = __builtin_amdgcn_wmma_f32_16x16x4_f32(false, a[0], false, b0, (short)0, acc, false, false);
    acc = __builtin_amdgcn_wmma_f32_16x16x4_f32(false, a[1], false, b1, (short)0, acc, false, false);
    acc = __builtin_amdgcn_wmma_f32_16x16x4_f32(false, a[2], false, b2, (short)0, acc, false, false);
    acc = __builtin_amdgcn_wmma_f32_16x16x4_f32(false, a[3], false, b3, (short)0, acc, false, false);
    return acc;
}

// ---------------------------------------------------------------------------
// Kernel 1: blocks 0..NBLK-1 : per-block composite via chained fp32 WMMA.
//           blocks NBLK..+15 : gold-path partial sums (overlapped in-grid).
// diag(e)*expT is formed on the A operand: one v_exp per lane per step; the
// running product is passed straight back as B. Rescale via readfirstlane of
// the (0,0) entry (strictly positive), inverse deferred into the next A'.
// ---------------------------------------------------------------------------
__global__ __launch_bounds__(32) void crf_scan_gold(const float* __restrict__ feats,
                                                    const int*   __restrict__ tags,
                                                    const float* __restrict__ trans,
                                                    float* __restrict__ ws) {
    const int lane = threadIdx.x;
    const int lo16 = lane & 15;
    const int hi   = lane >> 4;

    if (blockIdx.x >= NBLK) {
        const int gb = blockIdx.x - NBLK;
        const int ch = SEQ_LEN / GOLD_BLKS;            // 2048
        float acc = 0.0f;
        for (int s = gb * ch + lane; s < (gb + 1) * ch; s += 32) {
            const int next = tags[s];
            const int prev = (s == 0) ? START_TAG : tags[s - 1];
            acc += trans[next * 16 + prev] + feats[s * 16 + next];
        }
        acc = wave_sum32(acc);                         // fixed order -> deterministic
        if (lane == 0) ws[WS_GOLD + gb] = acc;
        return;
    }

    const int b = blockIdx.x;
    const bool lo = (hi == 0);

    // expT in A-operand layout: a[kk] holds k = 4kk + 2*hi + {0,1}, row m = lo16
    v2f a[4];
    #pragma unroll
    for (int kk = 0; kk < 4; ++kk) {
        const int k0 = 4 * kk + 2 * hi;
        a[kk].x = __expf(trans[lo16 * 16 + k0 + 0]);   // exp(-10000) -> 0
        a[kk].y = __expf(trans[lo16 * 16 + k0 + 1]);
    }

    // Running product in C/D layout: c[r] = element (m = r + 8*hi, n = lo16). Init = I.
    float c[8];
    #pragma unroll
    for (int r = 0; r < 8; ++r) c[r] = ((r + 8 * hi) == lo16) ? 1.0f : 0.0f;

    float logscale2 = 0.0f;
    float invPrev   = 1.0f;                            // uniform (scalar regs)
    const int s0 = b * LSTEPS;

    for (int s = s0; s < s0 + LSTEPS; ++s) {
        // one emission per lane: row factor for A rows m = lo16 (coalesced b32)
        const float f = feats[s * 16 + lo16];
        const float e = __expf(f) * invPrev;           // off the dependency chain

        v2f ap[4];                                     // A' = diag(e) * expT / m_prev
        #pragma unroll
        for (int kk = 0; kk < 4; ++kk) { ap[kk].x = a[kk].x * e; ap[kk].y = a[kk].y * e; }

        v2f b0, b1, b2, b3;
        make_b(c, lo, b0, b1, b2, b3);
        v8f acc = wmma16(ap, b0, b1, b2, b3);
        #pragma unroll
        for (int r = 0; r < 8; ++r) c[r] = acc[r];

        const float m = rfl_f(c[0]);                   // (0,0) entry, > 0, uniform
        invPrev = __builtin_amdgcn_rcpf(m);            // scalar rcp
        logscale2 += __log2f(m);                       // scalar log + scalar add
    }

    #pragma unroll
    for (int r = 0; r < 8; ++r)
        ws[WS_MAT + b * 256 + (r + 8 * hi) * 16 + lo16] = c[r] * invPrev;
    if (lane == 0) ws[WS_SCALE + b] = logscale2 * LN2F;
}

// ---------------------------------------------------------------------------
// Kernel 2: one 1024-thread workgroup. 32 waves each fold FOLD=16 block
// composites into a group composite kept in LDS; after the barrier, wave 0
// folds the 32 group composites into the init vector and finalizes.
// ---------------------------------------------------------------------------
__global__ __launch_bounds__(1024) void crf_fold_combine(const float* __restrict__ trans,
                                                         const int*   __restrict__ tags,
                                                         const float* __restrict__ ws,
                                                         float* __restrict__ out) {
    __shared__ float gm[NGRP][256];
    __shared__ float gs[NGRP];
    __shared__ float vsh[16];

    const int tid  = threadIdx.x;
    const int g    = tid >> 5;                         // group == wave id
    const int lane = tid & 31;
    const int lo16 = lane & 15;
    const int hi   = lane >> 4;
    const bool lo  = (hi == 0);

    // ---- per-wave fold: Q = P_{16g+15} ... P_{16g} ----
    float c[8];
    #pragma unroll
    for (int r = 0; r < 8; ++r)
        c[r] = ws[WS_MAT + (g * FOLD) * 256 + (r + 8 * hi) * 16 + lo16];

    float logscale2 = 0.0f;
    float invPrev   = 1.0f;

    for (int t = 1; t < FOLD; ++t) {
        const float* Pm = ws + WS_MAT + (g * FOLD + t) * 256;
        v2f a[4];
        #pragma unroll
        for (int kk = 0; kk < 4; ++kk) {
            const float2 q = *(const float2*)(Pm + lo16 * 16 + 4 * kk + 2 * hi);
            a[kk].x = q.x * invPrev;                   // deferred inverse folded into A
            a[kk].y = q.y * invPrev;
        }
        v2f b0, b1, b2, b3;
        make_b(c, lo, b0, b1, b2, b3);
        v8f acc = wmma16(a, b0, b1, b2, b3);
        #pragma unroll
        for (int r = 0; r < 8; ++r) c[r] = acc[r];

        const float m = rfl_f(c[0]);
        invPrev = __builtin_amdgcn_rcpf(m);
        logscale2 += __log2f(m);
    }

    #pragma unroll
    for (int r = 0; r < 8; ++r)
        gm[g][(r + 8 * hi) * 16 + lo16] = c[r] * invPrev;

    float sv = (lane < FOLD) ? ws[WS_SCALE + g * FOLD + lane] : 0.0f;
    sv = wave_sum32(sv);                               // fixed order
    if (lane == 0) gs[g] = sv + logscale2 * LN2F;

    __syncthreads();

    // ---- wave 0: fold group composites into the init vector ----
    if (tid < 32) {
        const int row = lo16;
        float v[8];                                    // v[j] = state[8*hi + j]
        #pragma unroll
        for (int j = 0; j < 8; ++j) v[j] = ((8 * hi + j) == START_TAG) ? 1.0f : 0.0f;
        float total = 0.0f;

        for (int b = 0; b < NGRP; ++b) {
            const float* Pr = &gm[b][row * 16 + 8 * hi];
            float4 q0 = *(const float4*)(Pr);
            float4 q1 = *(const float4*)(Pr + 4);
            float p01 = q0.x*v[0] + q0.y*v[1];
            float p23 = q0.z*v[2] + q0.w*v[3];
            float p45 = q1.x*v[4] + q1.y*v[5];
            float p67 = q1.z*v[6] + q1.w*v[7];
            float part = (p01 + p23) + (p45 + p67);
            float y = part + permx16_f(part);          // full row dot (both halves)

            const float m = rfl_f(y);                  // row-0 dot, > 0, uniform
            total += __log2f(m) * LN2F + gs[b];
            const float inv = __builtin_amdgcn_rcpf(m);

            // single-wave LDS republish: DS ops are in-order within a wave
            if (hi == 0) vsh[row] = y * inv;
            float4 r0 = *(const float4*)(&vsh[8 * hi]);
            float4 r1 = *(const float4*)(&vsh[8 * hi + 4]);
            v[0]=r0.x; v[1]=r0.y; v[2]=r0.z; v[3]=r0.w;
            v[4]=r1.x; v[5]=r1.y; v[6]=r1.z; v[7]=r1.w;
        }

        float ps = 0.0f;
        #pragma unroll
        for (int j = 0; j < 8; ++j) ps += v[j] * __expf(trans[STOP_TAG * 16 + 8 * hi + j]);
        const float dot = ps + permx16_f(ps);

        if (lane == 0) {
            float gold = trans[STOP_TAG * 16 + tags[SEQ_LEN - 1]];
            for (int q = 0; q < GOLD_BLKS; ++q) gold += ws[WS_GOLD + q];   // fixed order
            const float forward = total + __log2f(dot) * LN2F;
            out[0] = (forward - gold) / (float)SEQ_LEN;
        }
    }
}

// ---------------------------------------------------------------------------
extern "C" void kernel_launch(void* const* d_in, const int* in_sizes, int n_in,
                              void* d_out, int out_size, void* d_ws, size_t ws_size,
                              hipStream_t stream) {
    const float* feats = (const float*)d_in[0];   // (32768, 16) f32
    const int*   tags  = (const int*)d_in[1];     // (32768,)    i32
    const float* trans = (const float*)d_in[2];   // (16, 16)    f32
    float* out = (float*)d_out;
    float* ws  = (float*)d_ws;

    crf_scan_gold<<<NBLK + GOLD_BLKS, 32, 0, stream>>>(feats, tags, trans, ws);
    crf_fold_combine<<<1, 1024, 0, stream>>>(trans, tags, ws, out);
}